// NonLocalBlock_13511967113795
// MI455X (gfx1250) — compile-verified
//
#include <hip/hip_runtime.h>
#include <hip/hip_bf16.h>

// ---------------------------------------------------------------------------
// NonLocalBlock (no softmax) — algebraically collapsed:
//   G[b]  = X Xt                      (256x256 Gram, WMMA f32 16x16x4)
//   P[b]  = phi_w @ G                 (128x256, VALU)
//   M[b]  = P @ g_w^T                 (128x128, VALU)
//   R[b]  = w_w @ M^T                 (256x128, VALU)
//   E[b]  = R @ theta_w / N           (256x256, VALU)
//   z[b]  = E X + X                   (WMMA f32 16x16x4, fused residual)
//   out   = InstanceNorm2d(z)         (in-place on d_out)
// ---------------------------------------------------------------------------

typedef __attribute__((ext_vector_type(2))) float v2f;
typedef __attribute__((ext_vector_type(8))) float v8f;

#define NB     8
#define CCH    256
#define ICH    128
#define NSP    4096      // H*W
#define EPS    1e-5f

#define LDSA   36        // padded LDS row stride for 32-wide K tiles
#define LDSX   132       // padded LDS row stride for 128-wide N tiles

// ---------------------------------------------------------------------------
// Kernel 1: per-batch Gram matrix G = X @ X^T  (C x C), X is [C, N] row-major.
// Grid: (4, 4, B); 64x64 tile per workgroup; 8 waves (4x2), 2 accs per wave.
// ---------------------------------------------------------------------------
__global__ __launch_bounds__(256) void gram_kernel(const float* __restrict__ x,
                                                   float* __restrict__ G) {
    const int b  = blockIdx.z;
    const int i0 = blockIdx.y * 64;
    const int j0 = blockIdx.x * 64;
    const float* __restrict__ X = x + (size_t)b * CCH * NSP;

    __shared__ float Al[64 * LDSA];
    __shared__ float Bl[64 * LDSA];

    const int tid  = threadIdx.x;
    const int w    = tid >> 5;
    const int lane = tid & 31;
    const int wr   = w >> 1;          // 0..3 -> 16-row band
    const int wc   = w & 1;           // 0..1 -> 32-col band
    const int hl   = lane & 15;
    const int hi   = lane >> 4;

    v8f acc0 = {};
    v8f acc1 = {};

    for (int kk = 0; kk < NSP; kk += 32) {
        __syncthreads();
        #pragma unroll
        for (int it = 0; it < 2; ++it) {
            const int idx = tid + it * 256;       // 0..511
            const int r   = idx >> 3;             // row 0..63
            const int c4  = (idx & 7) << 2;       // col 0..28 step 4
            const float4 av = *(const float4*)(X + (size_t)(i0 + r) * NSP + kk + c4);
            *(float4*)(&Al[r * LDSA + c4]) = av;
            const float4 bv = *(const float4*)(X + (size_t)(j0 + r) * NSP + kk + c4);
            *(float4*)(&Bl[r * LDSA + c4]) = bv;
        }
        __syncthreads();

        #pragma unroll
        for (int k = 0; k < 32; k += 4) {
            const int ks = k + 2 * hi;            // lanes 0-15: K=k,k+1; 16-31: K=k+2,k+3
            v2f a, b0, b1;
            a.x  = Al[(wr * 16 + hl) * LDSA + ks];
            a.y  = Al[(wr * 16 + hl) * LDSA + ks + 1];
            b0.x = Bl[(wc * 32 + hl) * LDSA + ks];
            b0.y = Bl[(wc * 32 + hl) * LDSA + ks + 1];
            b1.x = Bl[(wc * 32 + 16 + hl) * LDSA + ks];
            b1.y = Bl[(wc * 32 + 16 + hl) * LDSA + ks + 1];
            acc0 = __builtin_amdgcn_wmma_f32_16x16x4_f32(false, a, false, b0,
                                                         (short)0, acc0, false, false);
            acc1 = __builtin_amdgcn_wmma_f32_16x16x4_f32(false, a, false, b1,
                                                         (short)0, acc1, false, false);
        }
    }

    float* __restrict__ Gb = G + (size_t)b * CCH * CCH;
    #pragma unroll
    for (int r = 0; r < 8; ++r) {
        const int row = i0 + wr * 16 + r + hi * 8;
        const int col = j0 + wc * 32 + hl;
        Gb[(size_t)row * CCH + col]      = acc0[r];
        Gb[(size_t)row * CCH + col + 16] = acc1[r];
    }
}

// ---------------------------------------------------------------------------
// Kernel 2: generic small batched GEMM  C = scale * A @ B(^T)
// One thread per output element; matrices are tiny and L2-resident.
// ---------------------------------------------------------------------------
__global__ __launch_bounds__(256) void gemm_small(const float* __restrict__ A,
                                                  const float* __restrict__ B,
                                                  float* __restrict__ Cout,
                                                  int M, int Nn, int K,
                                                  int lda, int ldb, int ldc,
                                                  long strideA, long strideB, long strideC,
                                                  int transB, float scale) {
    const int b   = blockIdx.y;
    const int idx = blockIdx.x * blockDim.x + threadIdx.x;
    if (idx >= M * Nn) return;
    const int m = idx / Nn;
    const int n = idx % Nn;

    const float* __restrict__ Ar = A + (size_t)b * strideA + (size_t)m * lda;
    const float* __restrict__ Bb = B + (size_t)b * strideB;

    float s = 0.0f;
    if (transB) {
        const float* __restrict__ Br = Bb + (size_t)n * ldb;
        #pragma unroll 8
        for (int k = 0; k < K; ++k) s = fmaf(Ar[k], Br[k], s);
    } else {
        #pragma unroll 8
        for (int k = 0; k < K; ++k) s = fmaf(Ar[k], Bb[(size_t)k * ldb + n], s);
    }
    Cout[(size_t)b * strideC + (size_t)m * ldc + n] = s * scale;
}

// ---------------------------------------------------------------------------
// Kernel 3: z = E @ X + X  (fused residual), z -> d_out.
// Grid: (N/128, C/64, B). 64x128 tile; 8 waves (4x2), 4 accs per wave.
// ---------------------------------------------------------------------------
__global__ __launch_bounds__(256) void z_kernel(const float* __restrict__ x,
                                                const float* __restrict__ E,
                                                float* __restrict__ z) {
    const int b  = blockIdx.z;
    const int i0 = blockIdx.y * 64;     // output channel rows
    const int n0 = blockIdx.x * 128;    // spatial cols
    const float* __restrict__ Xb = x + (size_t)b * CCH * NSP;
    const float* __restrict__ Eb = E + (size_t)b * CCH * CCH;

    __shared__ float El[64 * LDSA];     // 64 rows x 32 K
    __shared__ float Xl[32 * LDSX];     // 32 K x 128 cols

    const int tid  = threadIdx.x;
    const int w    = tid >> 5;
    const int lane = tid & 31;
    const int wr   = w >> 1;            // 16-row band
    const int wc   = w & 1;             // 64-col band
    const int hl   = lane & 15;
    const int hi   = lane >> 4;

    v8f acc[4] = {v8f{}, v8f{}, v8f{}, v8f{}};

    for (int kk = 0; kk < CCH; kk += 32) {
        __syncthreads();
        #pragma unroll
        for (int it = 0; it < 2; ++it) {            // E tile: 64x32
            const int idx = tid + it * 256;
            const int r   = idx >> 3;
            const int c4  = (idx & 7) << 2;
            const float4 ev = *(const float4*)(Eb + (size_t)(i0 + r) * CCH + kk + c4);
            *(float4*)(&El[r * LDSA + c4]) = ev;
        }
        #pragma unroll
        for (int it = 0; it < 4; ++it) {            // X tile: 32x128, coalesced in n
            const int idx = tid + it * 256;         // 0..1023 float4s
            const int r   = idx >> 5;               // K row 0..31
            const int c4  = (idx & 31) << 2;        // col 0..124 step 4
            const float4 xv = *(const float4*)(Xb + (size_t)(kk + r) * NSP + n0 + c4);
            *(float4*)(&Xl[r * LDSX + c4]) = xv;
        }
        __syncthreads();

        #pragma unroll
        for (int k = 0; k < 32; k += 4) {
            const int ks = k + 2 * hi;
            v2f a;
            a.x = El[(wr * 16 + hl) * LDSA + ks];
            a.y = El[(wr * 16 + hl) * LDSA + ks + 1];
            #pragma unroll
            for (int j = 0; j < 4; ++j) {
                v2f bf;
                const int n = wc * 64 + j * 16 + hl;
                bf.x = Xl[ks * LDSX + n];
                bf.y = Xl[(ks + 1) * LDSX + n];
                acc[j] = __builtin_amdgcn_wmma_f32_16x16x4_f32(false, a, false, bf,
                                                               (short)0, acc[j], false, false);
            }
        }
    }

    float* __restrict__ zb = z + (size_t)b * CCH * NSP;
    #pragma unroll
    for (int j = 0; j < 4; ++j) {
        #pragma unroll
        for (int r = 0; r < 8; ++r) {
            const int row = i0 + wr * 16 + r + hi * 8;
            const int col = n0 + wc * 64 + j * 16 + hl;
            zb[(size_t)row * NSP + col] = acc[j][r] + Xb[(size_t)row * NSP + col];
        }
    }
}

// ---------------------------------------------------------------------------
// Kernel 4: InstanceNorm2d(affine=False) in place on z (d_out).
// One 256-thread block per (b,c) row of 4096 values, held in registers.
// ---------------------------------------------------------------------------
__global__ __launch_bounds__(256) void instnorm_kernel(float* __restrict__ z) {
    const size_t base = (size_t)blockIdx.x * NSP;
    const int tid = threadIdx.x;

    float v[16];
    float s = 0.0f, s2 = 0.0f;
    #pragma unroll
    for (int i = 0; i < 16; ++i) {
        v[i] = z[base + tid + i * 256];
        s  += v[i];
        s2 = fmaf(v[i], v[i], s2);
    }

    __shared__ float rs[256];
    __shared__ float rq[256];
    rs[tid] = s;
    rq[tid] = s2;
    __syncthreads();
    #pragma unroll
    for (int off = 128; off > 0; off >>= 1) {
        if (tid < off) { rs[tid] += rs[tid + off]; rq[tid] += rq[tid + off]; }
        __syncthreads();
    }

    const float mu   = rs[0] * (1.0f / NSP);
    const float var  = rq[0] * (1.0f / NSP) - mu * mu;
    const float rstd = rsqrtf(var + EPS);

    #pragma unroll
    for (int i = 0; i < 16; ++i)
        z[base + tid + i * 256] = (v[i] - mu) * rstd;
}

// ---------------------------------------------------------------------------
extern "C" void kernel_launch(void* const* d_in, const int* in_sizes, int n_in,
                              void* d_out, int out_size, void* d_ws, size_t ws_size,
                              hipStream_t stream) {
    const float* x       = (const float*)d_in[0];
    const float* g_w     = (const float*)d_in[1];
    const float* theta_w = (const float*)d_in[2];
    const float* phi_w   = (const float*)d_in[3];
    const float* w_w     = (const float*)d_in[4];
    float* out = (float*)d_out;
    float* ws  = (float*)d_ws;

    // Scratch layout (floats): total 1,703,936 floats = 6.8 MB
    float* G  = ws;                         // B*256*256
    float* P  = G  + (size_t)NB * CCH * CCH;  // B*128*256
    float* Mm = P  + (size_t)NB * ICH * CCH;  // B*128*128
    float* R  = Mm + (size_t)NB * ICH * ICH;  // B*256*128
    float* E  = R  + (size_t)NB * CCH * ICH;  // B*256*256

    // 1) Gram: G[b] = X Xt
    gram_kernel<<<dim3(CCH / 64, CCH / 64, NB), 256, 0, stream>>>(x, G);

    // 2) tiny chain (all per-batch, L2-resident)
    // P = phi_w @ G            [128,256] = [128,256]x[256,256]
    gemm_small<<<dim3((ICH * CCH + 255) / 256, NB), 256, 0, stream>>>(
        phi_w, G, P, ICH, CCH, CCH, CCH, CCH, CCH,
        0L, (long)CCH * CCH, (long)ICH * CCH, 0, 1.0f);
    // M = P @ g_w^T            [128,128] = [128,256]x[256,128]
    gemm_small<<<dim3((ICH * ICH + 255) / 256, NB), 256, 0, stream>>>(
        P, g_w, Mm, ICH, ICH, CCH, CCH, CCH, ICH,
        (long)ICH * CCH, 0L, (long)ICH * ICH, 1, 1.0f);
    // R = w_w @ M^T            [256,128] = [256,128]x[128,128]
    gemm_small<<<dim3((CCH * ICH + 255) / 256, NB), 256, 0, stream>>>(
        w_w, Mm, R, CCH, ICH, ICH, ICH, ICH, ICH,
        0L, (long)ICH * ICH, (long)CCH * ICH, 1, 1.0f);
    // E = (R @ theta_w) / N    [256,256] = [256,128]x[128,256]
    gemm_small<<<dim3((CCH * CCH + 255) / 256, NB), 256, 0, stream>>>(
        R, theta_w, E, CCH, CCH, ICH, ICH, CCH, CCH,
        (long)CCH * ICH, 0L, (long)CCH * CCH, 0, 1.0f / (float)NSP);

    // 3) z = E X + X  -> d_out
    z_kernel<<<dim3(NSP / 128, CCH / 64, NB), 256, 0, stream>>>(x, E, out);

    // 4) InstanceNorm in place on d_out
    instnorm_kernel<<<dim3(NB * CCH), 256, 0, stream>>>(out);
}